// GraphNeuralNetworkModel_57423712747658
// MI455X (gfx1250) — compile-verified
//
#include <hip/hip_runtime.h>
#include <hip/hip_bf16.h>

typedef __attribute__((ext_vector_type(2))) float v2f;
typedef __attribute__((ext_vector_type(8))) float v8f;

#define HID 128

// ---------------------------------------------------------------- utilities
__global__ void fill_i32(int* __restrict__ p, int v, int n) {
    int i = blockIdx.x * 256 + threadIdx.x;
    if (i < n) p[i] = v;
}

__global__ void copy_i32(const int* __restrict__ a, int* __restrict__ b, int n) {
    int i = blockIdx.x * 256 + threadIdx.x;
    if (i < n) b[i] = a[i];
}

// ---------------------------------------------------------------- CSR build
__global__ void count_kernel(const int* __restrict__ dst, int* __restrict__ cnt, int E) {
    int e = blockIdx.x * 256 + threadIdx.x;
    if (e < E) atomicAdd(&cnt[dst[e]], 1);
}

// per-block inclusive scan (Hillis-Steele) + block totals
__global__ void scan_block_kernel(const int* __restrict__ cnt, int* __restrict__ incl,
                                  int* __restrict__ blockSums, int N) {
    __shared__ int sh[256];
    int i = blockIdx.x * 256 + threadIdx.x;
    sh[threadIdx.x] = (i < N) ? cnt[i] : 0;
    __syncthreads();
    #pragma unroll
    for (int off = 1; off < 256; off <<= 1) {
        int t = (threadIdx.x >= off) ? sh[threadIdx.x - off] : 0;
        __syncthreads();
        sh[threadIdx.x] += t;
        __syncthreads();
    }
    if (i < N) incl[i] = sh[threadIdx.x];
    if (threadIdx.x == 255) blockSums[blockIdx.x] = sh[255];
}

// exclusive scan of block totals (nb ~ 196, trivial serial)
__global__ void scan_sums_kernel(int* __restrict__ blockSums, int nb) {
    if (threadIdx.x == 0 && blockIdx.x == 0) {
        int run = 0;
        for (int b = 0; b < nb; ++b) { int v = blockSums[b]; blockSums[b] = run; run += v; }
    }
}

// rowptr[i] = exclusive prefix; rowptr[N] = E
__global__ void scan_finalize_kernel(const int* __restrict__ cnt, const int* __restrict__ incl,
                                     const int* __restrict__ blockOffs, int* __restrict__ rowptr, int N) {
    int i = blockIdx.x * 256 + threadIdx.x;
    if (i < N) {
        int inc = incl[i] + blockOffs[blockIdx.x];
        rowptr[i] = inc - cnt[i];
        if (i == N - 1) rowptr[N] = inc;
    }
}

__global__ void place_kernel(const int* __restrict__ src, const int* __restrict__ dst,
                             int* __restrict__ fillpos, int* __restrict__ src_sorted, int E) {
    int e = blockIdx.x * 256 + threadIdx.x;
    if (e < E) {
        int p = atomicAdd(&fillpos[dst[e]], 1);
        src_sorted[p] = src[e];
    }
}

__global__ void dinv_kernel(const int* __restrict__ cnt, float* __restrict__ dinv, int N) {
    int i = blockIdx.x * 256 + threadIdx.x;
    if (i < N) dinv[i] = rsqrtf((float)cnt[i] + 1.0f);   // +1 self loop
}

// ------------------------------------------------- WMMA fp32 GEMM  out = A @ W (+b, relu)
// 64 rows per block; B fragments hoisted to VGPRs for the whole K, reused over 4 row tiles.
__global__ __launch_bounds__(256) void gemm128_wmma_kernel(
    const float* __restrict__ A, const float* __restrict__ W,
    const float* __restrict__ bias, float* __restrict__ out, int N, int flags)
{
    __shared__ float aTile[64][132];      // 33792 B, pitch 132 -> conflict-free ds_load_b64

    const int rowBase = blockIdx.x * 64;

    for (int t = threadIdx.x; t < 64 * HID; t += 256) {
        int r = t >> 7, c = t & 127;
        int row = rowBase + r;
        aTile[r][c] = (row < N) ? A[(size_t)row * HID + c] : 0.0f;
    }
    __syncthreads();

    const int wave = threadIdx.x >> 5;    // 8 waves -> 8 col tiles of 16
    const int lane = threadIdx.x & 31;
    const int m    = lane & 15;
    const int hsel = lane >> 4;
    const int cb   = wave * 16;

    // hoist all B fragments (W resident in WGP$/L2): 32 k-steps x 2 VGPRs
    v2f bfrag[32];
    #pragma unroll
    for (int kk = 0; kk < 32; ++kk) {
        int kt = kk * 4;
        bfrag[kk].x = W[(size_t)(kt + 2 * hsel)     * HID + cb + m];
        bfrag[kk].y = W[(size_t)(kt + 2 * hsel + 1) * HID + cb + m];
    }

    float bv = (flags & 2) ? bias[cb + m] : 0.0f;

    #pragma unroll
    for (int rt = 0; rt < 4; ++rt) {
        const int rb = rt * 16;
        v8f acc = {};
        #pragma unroll
        for (int kk = 0; kk < 32; ++kk) {
            int kt = kk * 4;
            v2f a;
            a.x = aTile[rb + m][kt + 2 * hsel];
            a.y = aTile[rb + m][kt + 2 * hsel + 1];
            acc = __builtin_amdgcn_wmma_f32_16x16x4_f32(
                      false, a, false, bfrag[kk], (short)0, acc, false, false);
        }
        #pragma unroll
        for (int v = 0; v < 8; ++v) {
            int row = rowBase + rb + v + 8 * hsel;
            if (row < N) {
                float val = acc[v] + bv;
                if (flags & 1) val = fmaxf(val, 0.0f);
                out[(size_t)row * HID + cb + m] = val;
            }
        }
    }
}

// ------------------------------------------------- fused GCN aggregation (gather, no atomics)
// h[i] = relu( sum_{j in N(i)} dinv_j*dinv_i*hw[j] + dinv_i^2*hw[i] + b )  (+ residual)
__global__ __launch_bounds__(256) void gcn_aggregate_kernel(
    const float* __restrict__ hw, const int* __restrict__ rowptr,
    const int* __restrict__ src_sorted, const float* __restrict__ dinv,
    const float* __restrict__ bias, float* __restrict__ h, int N, int residual)
{
    int i = blockIdx.x * 8 + (threadIdx.x >> 5);
    if (i >= N) return;
    int lane = threadIdx.x & 31;

    float di = dinv[i];
    float d2 = di * di;
    float4 self = ((const float4*)(hw + (size_t)i * HID))[lane];
    float ax = self.x * d2, ay = self.y * d2, az = self.z * d2, aw = self.w * d2;

    int lo = rowptr[i], hi = rowptr[i + 1];
    int e = lo;
    for (; e + 1 < hi; e += 2) {          // 2-way unroll to overlap gather latency
        int s0 = src_sorted[e], s1 = src_sorted[e + 1];
        float n0 = dinv[s0] * di, n1 = dinv[s1] * di;
        float4 v0 = ((const float4*)(hw + (size_t)s0 * HID))[lane];
        float4 v1 = ((const float4*)(hw + (size_t)s1 * HID))[lane];
        ax += v0.x * n0 + v1.x * n1;
        ay += v0.y * n0 + v1.y * n1;
        az += v0.z * n0 + v1.z * n1;
        aw += v0.w * n0 + v1.w * n1;
    }
    if (e < hi) {
        int s0 = src_sorted[e];
        float n0 = dinv[s0] * di;
        float4 v0 = ((const float4*)(hw + (size_t)s0 * HID))[lane];
        ax += v0.x * n0; ay += v0.y * n0; az += v0.z * n0; aw += v0.w * n0;
    }

    int f = lane * 4;
    float4 r;
    r.x = fmaxf(ax + bias[f + 0], 0.0f);
    r.y = fmaxf(ay + bias[f + 1], 0.0f);
    r.z = fmaxf(az + bias[f + 2], 0.0f);
    r.w = fmaxf(aw + bias[f + 3], 0.0f);

    float4* hp = (float4*)(h + (size_t)i * HID + f);
    if (residual) {
        float4 p = *hp;
        r.x += p.x; r.y += p.y; r.z += p.z; r.w += p.w;
    }
    *hp = r;
}

// ------------------------------------------------- pooling: batch is sorted -> binary search
__device__ __forceinline__ int lower_bound_i(const int* __restrict__ a, int n, int key) {
    int lo = 0, hi = n;
    while (lo < hi) { int mid = (lo + hi) >> 1; if (a[mid] < key) lo = mid + 1; else hi = mid; }
    return lo;
}

__global__ __launch_bounds__(256) void pool_kernel(
    const float* __restrict__ h, const int* __restrict__ batch,
    float* __restrict__ g_out, int N, int G)
{
    int g = blockIdx.x * 8 + (threadIdx.x >> 5);
    if (g >= G) return;
    int lane = threadIdx.x & 31;

    int lo = lower_bound_i(batch, N, g);
    int hi = lower_bound_i(batch, N, g + 1);

    float ax = 0.f, ay = 0.f, az = 0.f, aw = 0.f;
    for (int i = lo; i < hi; ++i) {
        float4 v = ((const float4*)(h + (size_t)i * HID))[lane];
        ax += v.x; ay += v.y; az += v.z; aw += v.w;
    }
    float inv = 1.0f / fmaxf((float)(hi - lo), 1.0f);
    float4 r = make_float4(ax * inv, ay * inv, az * inv, aw * inv);
    ((float4*)(g_out + (size_t)g * HID))[lane] = r;
}

// ------------------------------------------------- classifier (tiny)
__global__ void mlp1_kernel(const float* __restrict__ g, const float* __restrict__ Wc1,
                            const float* __restrict__ bc1, float* __restrict__ mid, int G)
{
    int idx = blockIdx.x * 256 + threadIdx.x;     // G*64
    if (idx >= G * 64) return;
    int gi = idx >> 6, j = idx & 63;
    const float* gr = g + (size_t)gi * HID;
    float s = bc1[j];
    #pragma unroll 8
    for (int k = 0; k < HID; ++k) s += gr[k] * Wc1[k * 64 + j];
    mid[idx] = fmaxf(s, 0.0f);
}

__global__ void mlp2_kernel(const float* __restrict__ mid, const float* __restrict__ Wc2,
                            const float* __restrict__ bc2, float* __restrict__ out, int G)
{
    int idx = blockIdx.x * 256 + threadIdx.x;     // G*10
    if (idx >= G * 10) return;
    int gi = idx / 10, c = idx % 10;
    const float* mr = mid + (size_t)gi * 64;
    float s = bc2[c];
    #pragma unroll 8
    for (int j = 0; j < 64; ++j) s += mr[j] * Wc2[j * 10 + c];
    out[idx] = s;
}

// ---------------------------------------------------------------- driver
extern "C" void kernel_launch(void* const* d_in, const int* in_sizes, int n_in,
                              void* d_out, int out_size, void* d_ws, size_t ws_size,
                              hipStream_t stream)
{
    const float* x     = (const float*)d_in[0];
    const int*   eidx  = (const int*)d_in[1];
    const int*   batch = (const int*)d_in[2];
    const float* W_in  = (const float*)d_in[3];
    const float* b_in  = (const float*)d_in[4];
    const float* Ws    = (const float*)d_in[5];
    const float* bs    = (const float*)d_in[6];
    const float* Wc1   = (const float*)d_in[7];
    const float* bc1   = (const float*)d_in[8];
    const float* Wc2   = (const float*)d_in[9];
    const float* bc2   = (const float*)d_in[10];

    const int N = in_sizes[0] / HID;          // 50000
    const int E = in_sizes[1] / 2;            // 800000
    const int G = 256;
    const int LAYERS = 3;

    const int* src = eidx;
    const int* dst = eidx + E;

    // workspace layout, each region padded to 64 elements (256B) for float4 alignment
    auto pad64 = [](size_t n) { return (n + 63) & ~(size_t)63; };
    char* wsb = (char*)d_ws;
    size_t off = 0;
    auto alloc = [&](size_t elems) { void* p = wsb + off * 4; off += pad64(elems); return p; };

    float* dinv       = (float*)alloc(N);
    float* h          = (float*)alloc((size_t)N * HID);
    float* hw         = (float*)alloc((size_t)N * HID);
    int*   cnt        = (int*)  alloc(N);
    int*   rowptr     = (int*)  alloc(N + 1);
    int*   incl       = (int*)  alloc(N);
    int*   blockSums  = (int*)  alloc(1024);
    int*   fillpos    = (int*)  alloc(N);
    int*   src_sorted = (int*)  alloc(E);
    float* pool       = (float*)alloc((size_t)G * HID);
    float* mid        = (float*)alloc((size_t)G * 64);

    const int B = 256;
    auto cdiv = [](int a, int b) { return (a + b - 1) / b; };
    const int nb = cdiv(N, B);

    // ---- CSR build (once per call) + dinv
    fill_i32<<<cdiv(N, B), B, 0, stream>>>(cnt, 0, N);
    count_kernel<<<cdiv(E, B), B, 0, stream>>>(dst, cnt, E);
    scan_block_kernel<<<nb, B, 0, stream>>>(cnt, incl, blockSums, N);
    scan_sums_kernel<<<1, B, 0, stream>>>(blockSums, nb);
    scan_finalize_kernel<<<nb, B, 0, stream>>>(cnt, incl, blockSums, rowptr, N);
    copy_i32<<<cdiv(N, B), B, 0, stream>>>(rowptr, fillpos, N);
    place_kernel<<<cdiv(E, B), B, 0, stream>>>(src, dst, fillpos, src_sorted, E);
    dinv_kernel<<<cdiv(N, B), B, 0, stream>>>(cnt, dinv, N);

    // ---- input layer: h = relu(x @ W_in + b_in)
    gemm128_wmma_kernel<<<cdiv(N, 64), B, 0, stream>>>(x, W_in, b_in, h, N, 3);

    // ---- GCN layers (GEMM + fused gather/bias/relu/residual)
    for (int l = 0; l < LAYERS; ++l) {
        const float* Wl = Ws + (size_t)l * HID * HID;
        const float* bl = bs + (size_t)l * HID;
        gemm128_wmma_kernel<<<cdiv(N, 64), B, 0, stream>>>(h, Wl, nullptr, hw, N, 0);
        gcn_aggregate_kernel<<<cdiv(N, 8), B, 0, stream>>>(hw, rowptr, src_sorted, dinv,
                                                           bl, h, N, l > 0 ? 1 : 0);
    }

    // ---- global mean pool (sorted batch, no atomics)
    pool_kernel<<<cdiv(G, 8), B, 0, stream>>>(h, batch, pool, N, G);

    // ---- classifier
    mlp1_kernel<<<cdiv(G * 64, B), B, 0, stream>>>(pool, Wc1, bc1, mid, G);
    mlp2_kernel<<<cdiv(G * 10, B), B, 0, stream>>>(mid, Wc2, bc2, (float*)d_out, G);
}